// SwapPredGnn_25494925869555
// MI455X (gfx1250) — compile-verified
//
#include <hip/hip_runtime.h>
#include <hip/hip_bf16.h>

// ---------------------------------------------------------------------------
// SwapPredGnn: 3x GCNConv (sym-norm, self-loops) + global_sort_pool(k=30)
// MI455X / gfx1250, wave32, fp32 WMMA (V_WMMA_F32_16X16X4_F32)
//
// Fusion plan (bandwidth-first, scatter is roofline-dominant):
//   deg/dinv computed once (shared by all 3 layers)
//   layer L:  GEMM (A staged in LDS with fused bias_{L-1}+ReLU)  -> bufA
//             zero(bufB)  [h_{L-1} in bufB is dead after the GEMM]
//             scatter over NE real edges + NN virtual self-loop edges -> bufB
//   b3 fused into sort-pool.
// ---------------------------------------------------------------------------

typedef float v2f __attribute__((ext_vector_type(2)));
typedef float v8f __attribute__((ext_vector_type(8)));

#define NN     100000     // nodes (= 6250 * 16, exact M tiling)
#define NE     1600000    // edges
#define NPG    100        // nodes per graph
#define NGRAPH 1000
#define POOLK  30

// --------------------------- degree / dinv ---------------------------------

__global__ void k_zero(float* __restrict__ p, int n) {
    int i = blockIdx.x * blockDim.x + threadIdx.x;
    if (i < n) p[i] = 0.0f;
}

__global__ void k_zero4(float4* __restrict__ p, int n4) {
    int i = blockIdx.x * blockDim.x + threadIdx.x;
    if (i < n4) p[i] = make_float4(0.f, 0.f, 0.f, 0.f);
}

__global__ void k_deg(const long long* __restrict__ dst, float* __restrict__ deg, int nE) {
    int e = blockIdx.x * blockDim.x + threadIdx.x;
    if (e < nE) unsafeAtomicAdd(&deg[(int)dst[e]], 1.0f);
}

// dinv = rsqrt(deg + 1)   (+1 is the self-loop; every node has one)
__global__ void k_dinv(float* __restrict__ deg, int n) {
    int i = blockIdx.x * blockDim.x + threadIdx.x;
    if (i < n) deg[i] = rsqrtf(deg[i] + 1.0f);
}

// --------------------------- fp32 WMMA GEMM --------------------------------
// Y[M,NCOLS] = act(X + preb)[M,128] @ W[128,NCOLS].
// One wave per 16x16 tile, 32 steps of V_WMMA_F32_16X16X4_F32.
// Block = 256 threads = 8 waves = 8 tiles = (128/NCOLS) m-tiles x (NCOLS/16)
// n-tiles. A rows staged in LDS (padded stride 132 -> conflict-free b64 reads),
// with the previous layer's bias+ReLU fused into the staging pass (once per
// element instead of 8x per redundant global read).

template <int NCOLS, bool PREACT>
__global__ __launch_bounds__(256) void k_gemm_wmma(
        const float* __restrict__ X, const float* __restrict__ W,
        const float* __restrict__ preb, float* __restrict__ Y) {
    constexpr int KDIM    = 128;
    constexpr int MTB     = 128 / NCOLS;       // m-tiles per block: 1 (N=128) / 2 (N=64)
    constexpr int NT      = NCOLS / 16;        // n-tiles: 8 / 4
    constexpr int ASTRIDE = KDIM + 4;          // 132: 132%64=4 -> conflict-free banks
    __shared__ float Alds[MTB * 16 * ASTRIDE];

    int tid  = threadIdx.x;
    int wave = tid >> 5;
    int lane = tid & 31;
    int m_base = blockIdx.x * (MTB * 16);

    // ---- stage A rows (fused pre-activation), coalesced float4 ----
    constexpr int ROW4 = KDIM / 4;                       // 32 float4 per row
    constexpr int TOT4 = MTB * 16 * ROW4;                // 512 / 1024
    const float4* __restrict__ Xr = (const float4*)(X + (size_t)m_base * KDIM);
#pragma unroll
    for (int i = tid; i < TOT4; i += 256) {
        float4 v = Xr[i];
        if (PREACT) {
            int c = (i & (ROW4 - 1)) << 2;               // input channel of .x
            v.x = fmaxf(v.x + preb[c + 0], 0.f);
            v.y = fmaxf(v.y + preb[c + 1], 0.f);
            v.z = fmaxf(v.z + preb[c + 2], 0.f);
            v.w = fmaxf(v.w + preb[c + 3], 0.f);
        }
        int r = i >> 5;                                  // i / ROW4
        int c4 = i & (ROW4 - 1);
        *(float4*)(Alds + r * ASTRIDE + (c4 << 2)) = v;  // 16B-aligned (132%4==0)
    }
    __syncthreads();

    // ---- per-wave 16x16 tile ----
    int mt_local = wave / NT;
    int nt       = wave - mt_local * NT;
    int lm   = lane & 15;
    int koff = (lane >> 4) << 1;                         // 0 | 2

    const float* __restrict__ arow = Alds + (mt_local * 16 + lm) * ASTRIDE;
    const float* __restrict__ bcol = W + (nt << 4) + lm;

    v8f c = {};
#pragma unroll 4
    for (int k = 0; k < KDIM; k += 4) {
        v2f a = *(const v2f*)(arow + k + koff);          // ds_load_b64, 8B aligned
        v2f b;
        b.x = bcol[(size_t)(k + koff    ) * NCOLS];
        b.y = bcol[(size_t)(k + koff + 1) * NCOLS];
        c = __builtin_amdgcn_wmma_f32_16x16x4_f32(
                /*neg_a=*/false, a, /*neg_b=*/false, b,
                /*c_mod=*/(short)0, c, /*reuse_a=*/false, /*reuse_b=*/false);
    }

    // C/D layout: VGPR v -> M = v + 8*(lane>=16), N = lane&15
    int mrow = m_base + mt_local * 16 + ((lane >> 4) << 3);
    float* __restrict__ yp = Y + (size_t)mrow * NCOLS + (nt << 4) + lm;
#pragma unroll
    for (int v = 0; v < 8; ++v) yp[(size_t)v * NCOLS] = c[v];
}

// ------- edge scatter: out[dst] += norm * xw[src]  (+ fused self-loops) -----
// Work items e in [0, NE) are real edges; e in [NE, NE+NN) are the self-loops
// (src=dst=e-NE, norm=dinv^2). One wave (C=128) / half-wave (C=64) per item;
// lanes cover channels in float4 -> coalesced gather + native f32 atomics.

__global__ void k_scatter(const float4* __restrict__ xw, const float* __restrict__ dinv,
                          const long long* __restrict__ src, const long long* __restrict__ dst,
                          float* __restrict__ out, int c4shift, int nTot) {
    int gid = blockIdx.x * blockDim.x + threadIdx.x;
    int e   = gid >> c4shift;
    if (e >= nTot) return;
    int c4  = gid & ((1 << c4shift) - 1);

    int s, d;
    float nrm;
    if (e < NE) {                       // wave-uniform branch
        s = (int)src[e];
        d = (int)dst[e];
        nrm = dinv[s] * dinv[d];
    } else {                            // fused self-loop
        s = d = e - NE;
        float dv = dinv[s];
        nrm = dv * dv;
    }
    float4 v  = xw[(s << c4shift) + c4];
    float* op = out + (((size_t)d << c4shift) << 2) + (c4 << 2);
    unsafeAtomicAdd(op + 0, v.x * nrm);
    unsafeAtomicAdd(op + 1, v.y * nrm);
    unsafeAtomicAdd(op + 2, v.z * nrm);
    unsafeAtomicAdd(op + 3, v.w * nrm);
}

// ----------------------- sort pool (b3 fused) -------------------------------
// Stable descending rank by last channel == jnp.argsort(-key) with index
// tie-break. One block (4 waves) per graph; O(100^2) rank in LDS.

__global__ void k_sortpool(const float* __restrict__ h, const float* __restrict__ b3,
                           float* __restrict__ out) {
    __shared__ float keys[NPG];
    __shared__ int   sel[POOLK];
    int g = blockIdx.x;
    int t = threadIdx.x;                       // blockDim = 128
    const float* hg = h + (size_t)g * NPG * 64;

    if (t < NPG) keys[t] = hg[t * 64 + 63] + b3[63];
    __syncthreads();

    if (t < NPG) {
        float ki = keys[t];
        int rank = 0;
#pragma unroll 4
        for (int j = 0; j < NPG; ++j) {
            float kj = keys[j];
            rank += (kj > ki) || (kj == ki && j < t);
        }
        if (rank < POOLK) sel[rank] = t;
    }
    __syncthreads();

    for (int idx = t; idx < POOLK * 64; idx += 128) {
        int r = idx >> 6, c = idx & 63;
        out[(size_t)g * (POOLK * 64) + idx] = hg[sel[r] * 64 + c] + b3[c];
    }
}

// ---------------------------------------------------------------------------

extern "C" void kernel_launch(void* const* d_in, const int* in_sizes, int n_in,
                              void* d_out, int out_size, void* d_ws, size_t ws_size,
                              hipStream_t stream) {
    const float*     x  = (const float*)d_in[0];        // [100000,128]
    const long long* ei = (const long long*)d_in[1];    // [2,1600000] int64
    // d_in[2] = batch: implicit (contiguous equal-size graphs)
    const float* W1 = (const float*)d_in[3];
    const float* b1 = (const float*)d_in[4];
    const float* W2 = (const float*)d_in[5];
    const float* b2 = (const float*)d_in[6];
    const float* W3 = (const float*)d_in[7];
    const float* b3 = (const float*)d_in[8];
    float* out = (float*)d_out;

    float* ws   = (float*)d_ws;
    float* dinv = ws;                        // 100000 floats (400000 B, 16B aligned)
    float* bufA = ws + NN;                   // xw      : 100000*128 floats
    float* bufB = bufA + (size_t)NN * 128;   // h (agg) : 100000*128 floats

    const long long* srcp = ei;              // edge_index[0]
    const long long* dstp = ei + NE;         // edge_index[1]

    // ---- normalization coefficients: once, shared by all three layers ----
    k_zero<<<(NN + 255) / 256, 256, 0, stream>>>(dinv, NN);
    k_deg<<<(NE + 255) / 256, 256, 0, stream>>>(dstp, dinv, NE);
    k_dinv<<<(NN + 255) / 256, 256, 0, stream>>>(dinv, NN);

    const int nTot = NE + NN;                // real edges + virtual self-loops

    // ---- layer 1: x @ W1, aggregate ----
    k_gemm_wmma<128, false><<<NN / 16, 256, 0, stream>>>(x, W1, nullptr, bufA);
    k_zero4<<<(NN * 32) / 256, 256, 0, stream>>>((float4*)bufB, NN * 32);
    k_scatter<<<((size_t)nTot * 32) / 256, 256, 0, stream>>>(
        (const float4*)bufA, dinv, srcp, dstp, bufB, 5, nTot);

    // ---- layer 2: relu(h1+b1) @ W2, aggregate ----
    k_gemm_wmma<128, true><<<NN / 16, 256, 0, stream>>>(bufB, W2, b1, bufA);
    k_zero4<<<(NN * 32) / 256, 256, 0, stream>>>((float4*)bufB, NN * 32);
    k_scatter<<<((size_t)nTot * 32) / 256, 256, 0, stream>>>(
        (const float4*)bufA, dinv, srcp, dstp, bufB, 5, nTot);

    // ---- layer 3: relu(h2+b2) @ W3 (N=64), aggregate ----
    k_gemm_wmma<64, true><<<NN / 32, 256, 0, stream>>>(bufB, W3, b2, bufA);
    k_zero4<<<(NN * 16) / 256, 256, 0, stream>>>((float4*)bufB, NN * 16);
    k_scatter<<<((size_t)nTot * 16) / 256, 256, 0, stream>>>(
        (const float4*)bufA, dinv, srcp, dstp, bufB, 4, nTot);

    // ---- global_sort_pool(k=30), b3 fused -> out [1000, 1920] ----
    k_sortpool<<<NGRAPH, 128, 0, stream>>>(bufB, b3, out);
}